// TopLayer_72636486910639
// MI455X (gfx1250) — compile-verified
//
#include <hip/hip_runtime.h>
#include <hip/hip_bf16.h>

typedef float v2f __attribute__((ext_vector_type(2)));
typedef float v4f __attribute__((ext_vector_type(4)));
typedef float v8f __attribute__((ext_vector_type(8)));
typedef __attribute__((__vector_size__(4 * sizeof(int)))) int v4i_t;

#define Bn   8
#define Tn   256
#define Dn   768
#define Vn   32000
#define Rn   32
#define Mn   (Bn * Tn)      // 2048
#define BEAM 64
#define PADTOK 1

#define NEG_INF (-__builtin_inff())

#if __has_builtin(__builtin_amdgcn_global_load_async_to_lds_b128) && \
    __has_builtin(__builtin_amdgcn_s_wait_asynccnt)
#define HAVE_ASYNC_LDS 1
#define GLB_CAST(p) ((__attribute__((address_space(1))) v4i_t*)(p))
#define LDS_CAST(p) ((__attribute__((address_space(3))) v4i_t*)(p))
#else
#define HAVE_ASYNC_LDS 0
#endif

// ---------------- workspace layout (bytes) ----------------
#define EMIS_OFF   ((size_t)0)                                     // Mn*Vn f32
#define BTOK_OFF   ((size_t)262144000)                             // Mn*64 i32
#define BEMIT_OFF  (BTOK_OFF + (size_t)Mn * 64 * 4)                // Mn*64 f32
#define TRANS_OFF  (BEMIT_OFF + (size_t)Mn * 64 * 4)               // Bn*(Tn-1)*4096 f32
#define ACC_OFF    (TRANS_OFF + (size_t)Bn * (Tn - 1) * 64 * 64 * 4)

__device__ __forceinline__ unsigned keyOf(float f) {
    unsigned u = __float_as_uint(f);
    return (u & 0x80000000u) ? ~u : (u | 0x80000000u);   // ascending-monotonic
}

// =====================================================================
// 0) zero accumulators
// =====================================================================
__global__ void init_acc(float* acc) {
    if (threadIdx.x < 4) acc[threadIdx.x] = 0.0f;
}

// =====================================================================
// 1) emissions GEMM: emis[m,n] = sum_k src[m,k]*Wp[n,k] + bp[n]
//    fp32 WMMA 16x16x4.  Block tile 128(M) x 256(N), 8 waves as 4x2,
//    wave tile 32x128 (2x8 WMMA subtiles).  K staged 16-wide via
//    double-buffered async-to-LDS copies (gfx1250 path).
// =====================================================================
__global__ __launch_bounds__(256) void gemm_emis(const float* __restrict__ src,
                                                 const float* __restrict__ Wp,
                                                 const float* __restrict__ bp,
                                                 float* __restrict__ emis) {
    __shared__ __align__(16) float As[2][128 * 20];  // stride 20: conflict-free + 16B rows
    __shared__ __align__(16) float Bs[2][256 * 20];

    const int tid   = threadIdx.x;
    const int lane  = tid & 31;
    const int wave  = tid >> 5;
    const int wm    = wave >> 1;        // 0..3  (M)
    const int wn    = wave & 1;         // 0..1  (N)
    const int m_wg  = blockIdx.y * 128;
    const int n_wg  = blockIdx.x * 256;
    const int l16   = lane & 15;
    const int lhalf = lane >> 4;

    v8f acc[2][8];
#pragma unroll
    for (int i = 0; i < 2; i++)
#pragma unroll
        for (int j = 0; j < 8; j++) {
            v8f z = {0.f, 0.f, 0.f, 0.f, 0.f, 0.f, 0.f, 0.f};
            acc[i][j] = z;
        }

    const int arow = tid >> 2;           // 0..63
    const int acol = (tid & 3) * 4;      // 0,4,8,12
    const float* aRow0 = src + (size_t)(m_wg + arow) * Dn + acol;
    const float* aRow1 = aRow0 + (size_t)64 * Dn;
    const float* wRow  = Wp + (size_t)(n_wg + tid) * Dn;

    auto compute = [&](int bi) {
#pragma unroll
        for (int ks = 0; ks < 4; ks++) {
            const int kk = ks * 4 + lhalf * 2;
            v2f a[2], bo[8];
#pragma unroll
            for (int im = 0; im < 2; im++) {
                const int m = wm * 32 + im * 16 + l16;
                a[im] = *(const v2f*)(&As[bi][m * 20 + kk]);
            }
#pragma unroll
            for (int jn = 0; jn < 8; jn++) {
                const int n = wn * 128 + jn * 16 + l16;
                bo[jn] = *(const v2f*)(&Bs[bi][n * 20 + kk]);
            }
#pragma unroll
            for (int im = 0; im < 2; im++)
#pragma unroll
                for (int jn = 0; jn < 8; jn++)
                    acc[im][jn] = __builtin_amdgcn_wmma_f32_16x16x4_f32(
                        false, a[im], false, bo[jn], (short)0, acc[im][jn], false, false);
        }
    };

#if HAVE_ASYNC_LDS
    auto stage = [&](int bi, int k0) {
        __builtin_amdgcn_global_load_async_to_lds_b128(
            GLB_CAST(aRow0 + k0), LDS_CAST(&As[bi][arow * 20 + acol]), 0, 0);
        __builtin_amdgcn_global_load_async_to_lds_b128(
            GLB_CAST(aRow1 + k0), LDS_CAST(&As[bi][(arow + 64) * 20 + acol]), 0, 0);
#pragma unroll
        for (int j = 0; j < 4; j++)
            __builtin_amdgcn_global_load_async_to_lds_b128(
                GLB_CAST(wRow + k0 + j * 4), LDS_CAST(&Bs[bi][tid * 20 + j * 4]), 0, 0);
    };

    stage(0, 0);
    int buf = 0;
    for (int k0 = 0; k0 < Dn - 16; k0 += 16) {
        stage(buf ^ 1, k0 + 16);                 // prefetch next K-chunk
        __builtin_amdgcn_s_wait_asynccnt(6);     // oldest 6 (current buffer) done
        __syncthreads();
        compute(buf);
        __syncthreads();                         // all readers done before overwrite
        buf ^= 1;
    }
    __builtin_amdgcn_s_wait_asynccnt(0);
    __syncthreads();
    compute(buf);
#else
    for (int k0 = 0; k0 < Dn; k0 += 16) {
        v4f av0 = *(const v4f*)(aRow0 + k0);
        *(v4f*)(&As[0][arow * 20 + acol]) = av0;
        v4f av1 = *(const v4f*)(aRow1 + k0);
        *(v4f*)(&As[0][(arow + 64) * 20 + acol]) = av1;
#pragma unroll
        for (int j = 0; j < 4; j++) {
            v4f bv = *(const v4f*)(wRow + k0 + j * 4);
            *(v4f*)(&Bs[0][tid * 20 + j * 4]) = bv;
        }
        __syncthreads();
        compute(0);
        __syncthreads();
    }
#endif

    // epilogue: D layout -> lane L: col = L%16, row = r + 8*(L/16)
#pragma unroll
    for (int jn = 0; jn < 8; jn++) {
        const int n = n_wg + wn * 128 + jn * 16 + l16;
        const float bias = bp[n];
#pragma unroll
        for (int im = 0; im < 2; im++) {
#pragma unroll
            for (int r = 0; r < 8; r++) {
                const int m = m_wg + wm * 32 + im * 16 + r + 8 * lhalf;
                emis[(size_t)m * Vn + n] = acc[im][jn][r] + bias;
            }
        }
    }
}

// =====================================================================
// 2) per-row: logsumexp (MLE partials) + radix-select top-63 excl. gold
//    Pass structure: [lse + byte3-hist] -> byte2 -> byte1 -> byte0 -> gather
//    All row sweeps vectorized to b128 loads.
// =====================================================================
__global__ __launch_bounds__(256) void row_select(const float* __restrict__ emis,
                                                  const int* __restrict__ tgt,
                                                  int* __restrict__ btok,
                                                  float* __restrict__ bemit,
                                                  float* acc) {
    const int row = blockIdx.x;
    const int tid = threadIdx.x;
    const float* e = emis + (size_t)row * Vn;
    const int gold = tgt[row];

    __shared__ float redm[256];
    __shared__ float reds[256];
    __shared__ unsigned hist[256];
    __shared__ unsigned s_prefix, s_k;
    __shared__ unsigned cntAbove, cntEq;

    // ---- fused pass: online logsumexp + byte-3 histogram ----
    hist[tid] = 0;
    if (tid == 0) { cntAbove = 0; cntEq = 0; }
    __syncthreads();

    float m = NEG_INF, s = 0.0f;
    for (int i0 = tid * 4; i0 < Vn; i0 += 1024) {
        v4f x4 = *(const v4f*)(e + i0);
#pragma unroll
        for (int c = 0; c < 4; c++) {
            float x = x4[c];
            if (x > m) { s = s * expf(m - x) + 1.0f; m = x; }
            else       { s += expf(x - m); }
            if (i0 + c != gold) atomicAdd(&hist[keyOf(x) >> 24], 1u);
        }
    }
    redm[tid] = m; reds[tid] = s;
    __syncthreads();
    for (int off = 128; off > 0; off >>= 1) {
        if (tid < off) {
            float m1 = redm[tid], s1 = reds[tid];
            float m2 = redm[tid + off], s2 = reds[tid + off];
            float mm = fmaxf(m1, m2);
            redm[tid] = mm;
            reds[tid] = s1 * expf(m1 - mm) + s2 * expf(m2 - mm);
        }
        __syncthreads();
    }
    const float logZ = redm[0] + logf(reds[0]);

    if (tid == 0) {
        float g = e[gold];
        if (gold != PADTOK) {
            atomicAdd(&acc[0], -(g - logZ));   // -tok_logp
            atomicAdd(&acc[1], 1.0f);          // n_valid
        }
        btok[(size_t)row * 64]  = gold;        // beam slot 0 = gold
        bemit[(size_t)row * 64] = g;
        // byte-3 selection (histogram complete: barrier above)
        unsigned cum = 0; int sel = 0; unsigned kk = 63;
        for (int bin = 255; bin >= 0; bin--) {
            unsigned h = hist[bin];
            if (cum + h >= kk) { sel = bin; break; }
            cum += h;
        }
        s_prefix = (unsigned)sel << 24;
        s_k = kk - cum;
    }
    __syncthreads();
    unsigned prefix = s_prefix, kRem = s_k;

    // ---- radix bytes 2..0 ----
    for (int byte = 2; byte >= 0; byte--) {
        __syncthreads();
        hist[tid] = 0;
        __syncthreads();
        const unsigned shift = (unsigned)byte * 8;
        const unsigned prefMask = 0xFFFFFFFFu << ((byte + 1) * 8);
        for (int i0 = tid * 4; i0 < Vn; i0 += 1024) {
            v4f x4 = *(const v4f*)(e + i0);
#pragma unroll
            for (int c = 0; c < 4; c++) {
                if (i0 + c == gold) continue;
                unsigned key = keyOf(x4[c]);
                if ((key & prefMask) == prefix)
                    atomicAdd(&hist[(key >> shift) & 255u], 1u);
            }
        }
        __syncthreads();
        if (tid == 0) {
            unsigned cum = 0; int sel = 0;
            for (int bin = 255; bin >= 0; bin--) {
                unsigned h = hist[bin];
                if (cum + h >= kRem) { sel = bin; break; }
                cum += h;
            }
            s_prefix = prefix | ((unsigned)sel << shift);
            s_k = kRem - cum;
        }
        __syncthreads();
        prefix = s_prefix; kRem = s_k;
    }

    // ---- gather: all > Tkey, plus kRem ties == Tkey ----
    const unsigned Tkey = prefix, q = kRem;
    for (int i0 = tid * 4; i0 < Vn; i0 += 1024) {
        v4f x4 = *(const v4f*)(e + i0);
#pragma unroll
        for (int c = 0; c < 4; c++) {
            if (i0 + c == gold) continue;
            float x = x4[c];
            unsigned key = keyOf(x);
            int slot = -1;
            if (key > Tkey) {
                slot = 1 + (int)atomicAdd(&cntAbove, 1u);
            } else if (key == Tkey) {
                unsigned eidx = atomicAdd(&cntEq, 1u);
                if (eidx < q) slot = 1 + (int)(63u - q) + (int)eidx;
            }
            if (slot >= 0) {
                btok[(size_t)row * 64 + slot]  = i0 + c;
                bemit[(size_t)row * 64 + slot] = x;
            }
        }
    }
}

// =====================================================================
// 3) all transition matrices: trans[b,t,k,l] = dot(E1[tokP[k]], E2[tokC[l]])
// =====================================================================
__global__ __launch_bounds__(256) void trans_all(const int* __restrict__ btok,
                                                 const float* __restrict__ E1,
                                                 const float* __restrict__ E2,
                                                 float* __restrict__ trans) {
    const int blk = blockIdx.x;                 // b*(Tn-1)+t
    const int b = blk / (Tn - 1);
    const int t = blk % (Tn - 1);
    const int tid = threadIdx.x;

    __shared__ float Ar[64 * 32];
    __shared__ float Br[64 * 33];               // padded: conflict-free

    const int* tokP = btok + ((size_t)(b * Tn + t)) * 64;
    const int* tokC = tokP + 64;
    for (int idx = tid; idx < 64 * 32; idx += 256) {
        int kk = idx >> 5, r = idx & 31;
        Ar[kk * 32 + r] = E1[(size_t)tokP[kk] * Rn + r];
        Br[kk * 33 + r] = E2[(size_t)tokC[kk] * Rn + r];
    }
    __syncthreads();

    const int l = tid & 63, kq = tid >> 6;
    float* outp = trans + (size_t)blk * 4096;
#pragma unroll 4
    for (int kk = kq * 16; kk < kq * 16 + 16; kk++) {
        float sum = 0.0f;
#pragma unroll
        for (int r = 0; r < 32; r++) sum += Ar[kk * 32 + r] * Br[l * 33 + r];
        outp[kk * 64 + l] = sum;
    }
}

// =====================================================================
// 4) sequential CRF scan per batch (software-pipelined trans loads)
// =====================================================================
__global__ __launch_bounds__(256) void crf_scan(const int* __restrict__ tgt,
                                                const float* __restrict__ bemit,
                                                const float* __restrict__ E1,
                                                const float* __restrict__ E2,
                                                const float* __restrict__ trans,
                                                float* acc) {
    const int b = blockIdx.x;
    const int tid = threadIdx.x;

    __shared__ float score[64];
    __shared__ float mred[4][64];
    __shared__ float sred[4][64];
    __shared__ float nred[256];

    // ---- numerator: one t per thread ----
    {
        const int t = tid;
        const int tok = tgt[b * Tn + t];
        float contrib = 0.0f;
        if (tok != PADTOK) {
            float eg = bemit[((size_t)(b * Tn + t)) * 64];   // gold emission
            float tg = 0.0f;
            if (t >= 1) {
                const int pt = tgt[b * Tn + t - 1];
#pragma unroll
                for (int r = 0; r < Rn; r++)
                    tg += E1[(size_t)pt * Rn + r] * E2[(size_t)tok * Rn + r];
            }
            contrib = eg + tg;
        }
        nred[tid] = contrib;
        __syncthreads();
        for (int off = 128; off > 0; off >>= 1) {
            if (tid < off) nred[tid] += nred[tid + off];
            __syncthreads();
        }
    }
    const float numerator = nred[0];

    const int l = tid & 63, kq = tid >> 6;
    if (tid < 64) score[tid] = bemit[((size_t)(b * Tn)) * 64 + tid];
    __syncthreads();

    const float* trbase = trans + (size_t)b * (Tn - 1) * 4096;

    // prefetch t=1 transition column
    float vnext[16];
#pragma unroll
    for (int j = 0; j < 16; j++) vnext[j] = trbase[(kq * 16 + j) * 64 + l];

    for (int t = 1; t < Tn; t++) {
        const bool valid = (tgt[b * Tn + t] != PADTOK);
        const float emit_l = bemit[((size_t)(b * Tn + t)) * 64 + l];

        float v[16];
        float mp = NEG_INF;
#pragma unroll
        for (int j = 0; j < 16; j++) {
            float x = score[kq * 16 + j] + vnext[j];
            v[j] = x;
            mp = fmaxf(mp, x);
        }
        // prefetch next step while reducing this one
        if (t < Tn - 1) {
            const float* trn = trbase + (size_t)t * 4096;
#pragma unroll
            for (int j = 0; j < 16; j++) vnext[j] = trn[(kq * 16 + j) * 64 + l];
        }

        mred[kq][l] = mp;
        __syncthreads();
        const float mfin = fmaxf(fmaxf(mred[0][l], mred[1][l]),
                                 fmaxf(mred[2][l], mred[3][l]));
        float sp = 0.0f;
#pragma unroll
        for (int j = 0; j < 16; j++) sp += expf(v[j] - mfin);
        sred[kq][l] = sp;
        __syncthreads();
        if (kq == 0) {
            float st = sred[0][l] + sred[1][l] + sred[2][l] + sred[3][l];
            float nxt = mfin + logf(st) + emit_l;
            if (valid) score[l] = nxt;
        }
        __syncthreads();
    }

    if (tid == 0) {
        float mm = NEG_INF;
        for (int i = 0; i < 64; i++) mm = fmaxf(mm, score[i]);
        float ss = 0.0f;
        for (int i = 0; i < 64; i++) ss += expf(score[i] - mm);
        const float den = mm + logf(ss);
        atomicAdd(&acc[2], den - numerator);   // batch_crf = -(num - den)
    }
}

// =====================================================================
// 5) finalize
// =====================================================================
__global__ void finalize(const float* acc, float* out) {
    if (threadIdx.x == 0) {
        out[0] = acc[0] / acc[1];   // mle_loss
        out[1] = acc[2] / acc[1];   // crf_loss
    }
}

extern "C" void kernel_launch(void* const* d_in, const int* in_sizes, int n_in,
                              void* d_out, int out_size, void* d_ws, size_t ws_size,
                              hipStream_t stream) {
    const float* src = (const float*)d_in[0];   // (8,256,768)
    const int*   tgt = (const int*)d_in[1];     // (8,256)
    const float* Wp  = (const float*)d_in[2];   // (32000,768)
    const float* bp  = (const float*)d_in[3];   // (32000,)
    const float* E1  = (const float*)d_in[4];   // (32000,32)
    const float* E2  = (const float*)d_in[5];   // (32000,32)
    float* out = (float*)d_out;

    char* ws = (char*)d_ws;
    float* emis  = (float*)(ws + EMIS_OFF);
    int*   btok  = (int*)  (ws + BTOK_OFF);
    float* bemit = (float*)(ws + BEMIT_OFF);
    float* trans = (float*)(ws + TRANS_OFF);
    float* acc   = (float*)(ws + ACC_OFF);

    init_acc<<<1, 32, 0, stream>>>(acc);
    gemm_emis<<<dim3(Vn / 256, Mn / 128), 256, 0, stream>>>(src, Wp, bp, emis);
    row_select<<<Mn, 256, 0, stream>>>(emis, tgt, btok, bemit, acc);
    trans_all<<<Bn * (Tn - 1), 256, 0, stream>>>(btok, E1, E2, trans);
    crf_scan<<<Bn, 256, 0, stream>>>(tgt, bemit, E1, E2, trans, acc);
    finalize<<<1, 32, 0, stream>>>(acc, out);
}